// GCN_25958782337847
// MI455X (gfx1250) — compile-verified
//
#include <hip/hip_runtime.h>
#include <hip/hip_bf16.h>
#include <cstdint>
#include <cstddef>

#define N_NODES 50000
#define N_EDGES 600000
#define D_IN 128
#define D_HID 128
#define D_OUT 32
#define N_GRAPHS 64

#define CDIV(a, b) (((a) + (b) - 1) / (b))

typedef __attribute__((ext_vector_type(2))) float v2f;
typedef __attribute__((ext_vector_type(8))) float v8f;

// ---------------------------------------------------------------------------
// Degree / normalization precompute
// ---------------------------------------------------------------------------
__global__ void k_init_deg(float* __restrict__ deg) {
    int n = blockIdx.x * blockDim.x + threadIdx.x;
    if (n < N_NODES) deg[n] = 1.0f;  // self-loop contribution
}

__global__ void k_accum_deg(const long long* __restrict__ ei, float* __restrict__ deg) {
    int e = blockIdx.x * blockDim.x + threadIdx.x;
    if (e < N_EDGES) {
        int d = (int)ei[N_EDGES + e];  // dst row of edge_index [2, E]
        unsafeAtomicAdd(&deg[d], 1.0f);
    }
}

__global__ void k_dinv(float* __restrict__ deg) {
    int n = blockIdx.x * blockDim.x + threadIdx.x;
    if (n < N_NODES) deg[n] = rsqrtf(deg[n]);  // deg >= 1 always
}

// Pack (src, dst, norm) into one 16B record per edge: single b128 load in scatter.
__global__ void k_pack_edges(const long long* __restrict__ ei, const float* __restrict__ dinv,
                             float4* __restrict__ ep) {
    int e = blockIdx.x * blockDim.x + threadIdx.x;
    if (e < N_EDGES) {
        int s = (int)ei[e];
        int d = (int)ei[N_EDGES + e];
        float w = dinv[s] * dinv[d];
        ep[e] = make_float4(__int_as_float(s), __int_as_float(d), w, 0.0f);
    }
}

// ---------------------------------------------------------------------------
// Weight pre-pack into B-fragment order:
//   Wp2[p*N + col] = { W[2p][col], W[2p+1][col] }   (float2, p in [0, K/2))
// Lane (col=lane&15 within an n-tile, half=lane>>4) reads pair p = k/2 + half
// as ONE global_load_b64 with an immediate offset.
// ---------------------------------------------------------------------------
template <int N, int K>
__global__ void k_pack_w(const float* __restrict__ W, float* __restrict__ Wp) {
    int t = blockIdx.x * blockDim.x + threadIdx.x;  // over (K/2)*N pairs
    if (t < (K / 2) * N) {
        int p = t / N;
        int col = t - p * N;
        Wp[2 * t]     = W[(2 * p) * N + col];
        Wp[2 * t + 1] = W[(2 * p + 1) * N + col];
    }
}

// ---------------------------------------------------------------------------
// WMMA f32 GEMM: C[M,N] = A[M,K] @ W[K,N] (+ bias[N] if non-null)
// One wave32 per 16x16 output tile, V_WMMA_F32_16X16X4_F32, K fully unrolled.
// A-frag (ISA 7.12.2): lanes 0-15 -> rows, v0/v1 = K {0,1}; lanes 16-31 = K {2,3}.
// B-frag striped like C/D: v0 holds row K=kgrp, v1 row K=kgrp+1, kgrp=2*(lane>=16).
// N, K compile-time -> all loop loads use immediate offsets (no address math).
// ---------------------------------------------------------------------------
template <int N, int K>
__global__ __launch_bounds__(128) void k_wmma_gemm(const float* __restrict__ A,
                                                   const float* __restrict__ Wp,
                                                   const float* __restrict__ bias,
                                                   float* __restrict__ C,
                                                   int tilesM) {
    constexpr int tilesN = N / 16;
    const int lane = threadIdx.x & 31;
    const int tile = blockIdx.x * 4 + (threadIdx.x >> 5);
    if (tile >= tilesM * tilesN) return;
    const int tm   = tile / tilesN;
    const int tn   = tile - tm * tilesN;
    const int l15  = lane & 15;
    const int half = lane >> 4;  // 0 or 1
    const int row  = tm * 16 + l15;
    const int col  = tn * 16 + l15;

    const float* __restrict__ ap = A + (size_t)row * K + half * 2;
    const v2f* __restrict__ wp =
        reinterpret_cast<const v2f*>(Wp) + (size_t)half * N + col;

    v8f acc = {};
#pragma unroll
    for (int k = 0; k < K; k += 4) {
        v2f a = *reinterpret_cast<const v2f*>(ap + k);  // b64, imm offset
        v2f b = wp[(k / 2) * N];                        // b64, imm offset
        acc = __builtin_amdgcn_wmma_f32_16x16x4_f32(
            /*neg_a=*/false, a, /*neg_b=*/false, b,
            /*c_mod=*/(short)0, acc, /*reuse_a=*/false, /*reuse_b=*/false);
    }

    const float bv = bias ? bias[col] : 0.0f;
    const int rbase = tm * 16 + (half << 3);  // +8 for upper half-lanes
#pragma unroll
    for (int i = 0; i < 8; ++i)
        C[(size_t)(rbase + i) * N + col] = acc[i] + bv;
}

// ---------------------------------------------------------------------------
// Aggregation seed: out[n,f] = bias[f] + dinv[n]^2 * hW[n,f]
// (self-loop message folded in analytically -> no atomics, no memset needed)
// ---------------------------------------------------------------------------
__global__ void k_seed(const float* __restrict__ hW, const float* __restrict__ dinv,
                       const float* __restrict__ bias, float* __restrict__ out) {
    int idx = blockIdx.x * blockDim.x + threadIdx.x;
    if (idx < N_NODES * D_HID) {
        int n = idx >> 7;
        int f = idx & 127;
        float di = dinv[n];
        out[idx] = bias[f] + hW[idx] * (di * di);
    }
}

// ---------------------------------------------------------------------------
// Edge scatter: one wave per edge, each lane a float4 feature chunk.
// out[dst] += norm[e] * hW[src]   (native f32 atomics)
// ---------------------------------------------------------------------------
__global__ void k_scatter(const float4* __restrict__ ep, const float* __restrict__ hW,
                          float* __restrict__ out) {
    int t = blockIdx.x * blockDim.x + threadIdx.x;
    if (t >= N_EDGES * 32) return;
    int e = t >> 5;
    int c = t & 31;  // 32 chunks of 4 floats = 128 features
    float4 er = ep[e];  // single b128 (broadcast across the wave)
    int s = __float_as_int(er.x);
    int d = __float_as_int(er.y);
    float w = er.z;
    const float4 v = *reinterpret_cast<const float4*>(hW + (size_t)s * D_HID + c * 4);
    float* op = out + (size_t)d * D_HID + c * 4;
    unsafeAtomicAdd(op + 0, v.x * w);
    unsafeAtomicAdd(op + 1, v.y * w);
    unsafeAtomicAdd(op + 2, v.z * w);
    unsafeAtomicAdd(op + 3, v.w * w);
}

// ---------------------------------------------------------------------------
// Global mean pool
// ---------------------------------------------------------------------------
__global__ void k_pool_zero(float* __restrict__ sums, float* __restrict__ counts) {
    int t = blockIdx.x * blockDim.x + threadIdx.x;
    if (t < N_GRAPHS * D_OUT) sums[t] = 0.0f;
    if (t < N_GRAPHS) counts[t] = 0.0f;
}

__global__ void k_pool_acc(const long long* __restrict__ batch, const float* __restrict__ h,
                           float* __restrict__ sums, float* __restrict__ counts) {
    int t = blockIdx.x * blockDim.x + threadIdx.x;
    if (t >= N_NODES * D_OUT) return;
    int n = t >> 5;  // D_OUT == 32
    int f = t & 31;
    int g = (int)batch[n];
    unsafeAtomicAdd(&sums[g * D_OUT + f], h[t]);
    if (f == 0) unsafeAtomicAdd(&counts[g], 1.0f);
}

__global__ void k_pool_fin(const float* __restrict__ sums, const float* __restrict__ counts,
                           float* __restrict__ out) {
    int t = blockIdx.x * blockDim.x + threadIdx.x;
    if (t < N_GRAPHS * D_OUT) out[t] = sums[t] / fmaxf(counts[t >> 5], 1.0f);
}

// ---------------------------------------------------------------------------
// Host-side orchestration
// ---------------------------------------------------------------------------
extern "C" void kernel_launch(void* const* d_in, const int* in_sizes, int n_in,
                              void* d_out, int out_size, void* d_ws, size_t ws_size,
                              hipStream_t stream) {
    (void)in_sizes; (void)n_in; (void)out_size; (void)ws_size;

    const float*     x     = (const float*)d_in[0];
    const long long* ei    = (const long long*)d_in[1];   // int64 [2, E]
    const long long* batch = (const long long*)d_in[2];   // int64 [N]
    const float* Ws[4] = {(const float*)d_in[3], (const float*)d_in[5],
                          (const float*)d_in[7], (const float*)d_in[9]};
    const float* bs[4] = {(const float*)d_in[4], (const float*)d_in[6],
                          (const float*)d_in[8], (const float*)d_in[10]};
    const float* fcW = (const float*)d_in[11];
    const float* fcb = (const float*)d_in[12];

    // Workspace carve-up (all 256B aligned)
    char* ws = (char*)d_ws;
    size_t off = 0;
    auto carve = [&](size_t bytes) -> float* {
        float* p = (float*)(ws + off);
        off = (off + bytes + 255) & ~(size_t)255;
        return p;
    };
    float*  bufA   = carve((size_t)N_NODES * D_HID * 4);    // 25.6 MB
    float*  bufB   = carve((size_t)N_NODES * D_HID * 4);    // 25.6 MB
    float*  bufC   = carve((size_t)N_NODES * D_HID * 4);    // 25.6 MB
    float*  dinv   = carve((size_t)N_NODES * 4);            // deg -> dinv in place
    float4* ep     = (float4*)carve((size_t)N_EDGES * 16);  // packed edges 9.6 MB
    float*  hfc    = carve((size_t)N_NODES * D_OUT * 4);
    float*  sums   = carve((size_t)N_GRAPHS * D_OUT * 4);
    float*  counts = carve((size_t)N_GRAPHS * 4);
    float*  Wp[4];
    for (int i = 0; i < 4; ++i) Wp[i] = carve((size_t)D_HID * D_HID * 4);
    float*  fcWp   = carve((size_t)D_HID * D_OUT * 4);

    // 1. Symmetric normalization + edge packing
    k_init_deg  <<<CDIV(N_NODES, 256), 256, 0, stream>>>(dinv);
    k_accum_deg <<<CDIV(N_EDGES, 256), 256, 0, stream>>>(ei, dinv);
    k_dinv      <<<CDIV(N_NODES, 256), 256, 0, stream>>>(dinv);
    k_pack_edges<<<CDIV(N_EDGES, 256), 256, 0, stream>>>(ei, dinv, ep);

    // 2. Weight pre-pack into fragment order
    for (int l = 0; l < 4; ++l)
        k_pack_w<D_HID, D_HID><<<CDIV(D_HID * D_HID / 2, 256), 256, 0, stream>>>(Ws[l], Wp[l]);
    k_pack_w<D_OUT, D_HID><<<CDIV(D_OUT * D_HID / 2, 256), 256, 0, stream>>>(fcW, fcWp);

    // 3. Four GCN layers (ping-pong through bufA/B/C)
    float* pool3[3] = {bufA, bufB, bufC};
    const float* cur = x;
    const int tilesM = N_NODES / 16;  // 3125
    for (int l = 0; l < 4; ++l) {
        float* hW = nullptr; float* hout = nullptr;
        for (int i = 0; i < 3; ++i) {
            if ((const float*)pool3[i] == cur) continue;
            if (!hW) hW = pool3[i];
            else if (!hout) hout = pool3[i];
        }
        k_wmma_gemm<D_HID, D_HID><<<CDIV(tilesM * (D_HID / 16), 4), 128, 0, stream>>>(
            cur, Wp[l], nullptr, hW, tilesM);
        k_seed   <<<CDIV(N_NODES * D_HID, 256), 256, 0, stream>>>(hW, dinv, bs[l], hout);
        k_scatter<<<CDIV(N_EDGES * 32, 256), 256, 0, stream>>>(ep, hW, hout);
        cur = hout;
    }

    // 4. Final FC (with bias fused)
    k_wmma_gemm<D_OUT, D_HID><<<CDIV(tilesM * (D_OUT / 16), 4), 128, 0, stream>>>(
        cur, fcWp, fcb, hfc, tilesM);

    // 5. Global mean pool
    k_pool_zero<<<CDIV(N_GRAPHS * D_OUT, 256), 256, 0, stream>>>(sums, counts);
    k_pool_acc <<<CDIV(N_NODES * D_OUT, 256), 256, 0, stream>>>(batch, hfc, sums, counts);
    k_pool_fin <<<CDIV(N_GRAPHS * D_OUT, 256), 256, 0, stream>>>(sums, counts, (float*)d_out);
}